// SambaBlock_37366215475758
// MI455X (gfx1250) — compile-verified
//
#include <hip/hip_runtime.h>

// SambaBlock single-step, MI455X (gfx1250, wave32).
// Roofline: hidden_states stream (256 MB r + 256 MB w) dominates -> ~22us @
// 23.3 TB/s. GEMMs (6.5 GFLOP fp32) run on V_WMMA_F32_16X16X4_F32 with
// multi-accumulator waves (independent WMMA chains, A-fragment reuse across
// N-tiles) so the matrix pipe never serializes on the C->C dependency.

#define BB       16
#define KK       512
#define DMODEL   256
#define DINNER   512
#define DSTATE   16
#define DTRANK   16
#define DCONV    4
#define NTOK     (BB * KK)          // 8192 independent tokens

typedef float v2f __attribute__((ext_vector_type(2)));
typedef float v8f __attribute__((ext_vector_type(8)));

__device__ __forceinline__ float sigmoidf_(float x) { return 1.0f / (1.0f + __expf(-x)); }
__device__ __forceinline__ float siluf_(float x)    { return x * sigmoidf_(x); }
__device__ __forceinline__ float softplusf_(float x) {
    return (x > 20.0f) ? x : log1pf(__expf(x));
}

// ---------------------------------------------------------------------------
// Generic fp32 WMMA GEMM:  C(M,N) = A(M,K) @ W(N,K)^T
//   - one wave computes a 16 x (16*NT) tile: NT independent accumulator
//     chains, one shared A fragment per k-step (A traffic / NT).
//   - A frag (16x4 f32): lanes 0-15: M=lane, K=k0+{0,1};
//                        lanes16-31: M=lane-16, K=k0+{2,3}  -> one float2 load
//   - B frag (4x16 f32): mirrored from row-major (N,K) weights
//   - C/D: 8 VGPRs, row = r + 8*(lane>=16), col = lane&15
// SPLIT=true (in_proj): cols [0,512) -> hist[t,3,:], cols [512,1024) -> res ws
// SPLIT=false:          plain row-major store with leading dim NOUT
// ---------------------------------------------------------------------------
template<int KDIM, int NOUT, int NT, bool SPLIT>
__global__ __launch_bounds__(256)
void gemm_wmma(const float* __restrict__ A,     // (NTOK, KDIM)
               const float* __restrict__ W,     // (NOUT, KDIM)
               float* __restrict__ dst0,        // SPLIT: histOut | else: Out
               float* __restrict__ dst1)        // SPLIT: resOut  | else: unused
{
    const int lane = threadIdx.x & 31;
    const int wave = threadIdx.x >> 5;
    const int half = lane >> 4;
    const int l15  = lane & 15;
    const int m0 = blockIdx.x * 16;
    const int n0 = (blockIdx.y * 8 + wave) * (16 * NT);

    const float* aPtr = A + (size_t)(m0 + l15) * KDIM + 2 * half;
    const float* bPtr[NT];
#pragma unroll
    for (int j = 0; j < NT; ++j)
        bPtr[j] = W + (size_t)(n0 + j * 16 + l15) * KDIM + 2 * half;

    v8f c[NT];
#pragma unroll
    for (int j = 0; j < NT; ++j) c[j] = v8f{0.f,0.f,0.f,0.f,0.f,0.f,0.f,0.f};

#pragma unroll 2
    for (int k0 = 0; k0 < KDIM; k0 += 4) {
        v2f a = *(const v2f*)(aPtr + k0);
#pragma unroll
        for (int j = 0; j < NT; ++j) {
            v2f b = *(const v2f*)(bPtr[j] + k0);
            c[j] = __builtin_amdgcn_wmma_f32_16x16x4_f32(false, a, false, b,
                                                         (short)0, c[j], false, false);
        }
    }

    if (SPLIT) {
        // NT*16 divides 512, so each j-tile is uniformly x-half or res-half
#pragma unroll
        for (int j = 0; j < NT; ++j) {
            const int col = n0 + j * 16 + l15;
            if (n0 + j * 16 < DINNER) {
#pragma unroll
                for (int r = 0; r < 8; ++r) {
                    int row = m0 + r + 8 * half;
                    dst0[(size_t)row * (DCONV * DINNER) + 3 * DINNER + col] = c[j][r];
                }
            } else {
#pragma unroll
                for (int r = 0; r < 8; ++r) {
                    int row = m0 + r + 8 * half;
                    dst1[(size_t)row * DINNER + (col - DINNER)] = c[j][r];
                }
            }
        }
    } else {
#pragma unroll
        for (int j = 0; j < NT; ++j) {
            const int col = n0 + j * 16 + l15;
#pragma unroll
            for (int r = 0; r < 8; ++r) {
                int row = m0 + r + 8 * half;
                dst0[(size_t)row * NOUT + col] = c[j][r];
            }
        }
    }
}

// ---------------------------------------------------------------------------
// Fused per-token: conv roll + depthwise conv + SiLU + LayerNorm + x_proj +
// dt + softplus + selective-scan step + y + gating. 1 block/token, 256 thr,
// 2 channels per thread. hidden_states streamed exactly once (r+w).
// ---------------------------------------------------------------------------
__global__ __launch_bounds__(256)
void samba_step(const float* __restrict__ hidden_in,    // (NTOK,512,16)
                const float* __restrict__ conv_hist_in, // (NTOK,4,512)
                const float* __restrict__ conv_w,       // (512,4)
                const float* __restrict__ conv_b,       // (512)
                const float* __restrict__ x_proj_w,     // (48,512)
                const float* __restrict__ dt_w,         // (512,16)
                const float* __restrict__ dt_b,         // (512)
                const float* __restrict__ A_log,        // (512,16)
                const float* __restrict__ Dp,           // (512)
                const float* __restrict__ ln_g,         // (512)
                const float* __restrict__ ln_b,         // (512)
                const int*   __restrict__ rate,         // scalar
                float* __restrict__ h_out,              // (NTOK,512,16)
                float* __restrict__ hist_out,           // (NTOK,4,512), l=3 prewritten
                float* __restrict__ res_gate)           // in: res, out: y*silu(res)
{
    const int t   = blockIdx.x;
    const int tid = threadIdx.x;
    const int d0  = tid * 2;

    __shared__ float su[DINNER];
    __shared__ float red1[256];
    __shared__ float red2[256];
    __shared__ float part[192];
    __shared__ float xdbl[48];

    const float* ch = conv_hist_in + (size_t)t * (DCONV * DINNER);
    float*       ho = hist_out     + (size_t)t * (DCONV * DINNER);

    // conv roll + depthwise conv + SiLU
    float s[2];
    float lsum = 0.f, lsq = 0.f;
#pragma unroll
    for (int i = 0; i < 2; ++i) {
        int d = d0 + i;
        float h1 = ch[1 * DINNER + d];
        float h2 = ch[2 * DINNER + d];
        float h3 = ch[3 * DINNER + d];
        float xv = ho[3 * DINNER + d];          // x from in_proj kernel
        ho[0 * DINNER + d] = h1;
        ho[1 * DINNER + d] = h2;
        ho[2 * DINNER + d] = h3;
        float acc = conv_b[d]
                  + h1 * conv_w[d * DCONV + 0]
                  + h2 * conv_w[d * DCONV + 1]
                  + h3 * conv_w[d * DCONV + 2]
                  + xv * conv_w[d * DCONV + 3];
        float sv = siluf_(acc);
        s[i] = sv;
        lsum += sv;
        lsq  += sv * sv;
    }

    // LayerNorm reduction over 512 channels
    red1[tid] = lsum; red2[tid] = lsq;
    __syncthreads();
    for (int off = 128; off > 0; off >>= 1) {
        if (tid < off) { red1[tid] += red1[tid + off]; red2[tid] += red2[tid + off]; }
        __syncthreads();
    }
    float mu  = red1[0] * (1.0f / DINNER);
    float var = red2[0] * (1.0f / DINNER) - mu * mu;
    float rs  = rsqrtf(var + 1e-5f);

    float u[2];
#pragma unroll
    for (int i = 0; i < 2; ++i) {
        int d = d0 + i;
        float uv = (s[i] - mu) * rs * ln_g[d] + ln_b[d];
        u[i]  = uv;
        su[d] = uv;
    }
    __syncthreads();

    // x_dbl = x_proj_w @ u : 48 outputs, 4 partial sums each
    if (tid < 192) {
        int j = tid >> 2, p = tid & 3;
        const float* wrow = x_proj_w + (size_t)j * DINNER + p * 128;
        const float* uu   = su + p * 128;
        float acc = 0.f;
#pragma unroll 8
        for (int d = 0; d < 128; ++d) acc += wrow[d] * uu[d];
        part[tid] = acc;
    }
    __syncthreads();
    if (tid < 48)
        xdbl[tid] = part[tid * 4] + part[tid * 4 + 1] + part[tid * 4 + 2] + part[tid * 4 + 3];
    __syncthreads();

    const float inv_rate = 1.0f / (float)rate[0];

    const float* hin  = hidden_in + (size_t)t * (DINNER * DSTATE);
    float*       hout = h_out     + (size_t)t * (DINNER * DSTATE);

    float gate[2];
#pragma unroll
    for (int i = 0; i < 2; ++i) {
        int d = d0 + i;
        // delta = softplus(dt_w @ x_dbl[0:16] + dt_b) / rate
        float acc = dt_b[d];
#pragma unroll
        for (int r = 0; r < DTRANK; ++r) acc += dt_w[d * DTRANK + r] * xdbl[r];
        float delta = softplusf_(acc) * inv_rate;
        float du = delta * u[i];

        const float* hr = hin  + (size_t)d * DSTATE;
        float*       hw = hout + (size_t)d * DSTATE;
        float y = 0.f;
#pragma unroll
        for (int n = 0; n < DSTATE; ++n) {
            float dA = __expf(-delta * __expf(A_log[d * DSTATE + n]));  // exp(dt*A), A=-exp(A_log)
            float hv = dA * hr[n] + du * xdbl[16 + n];                  // + dt*u*B
            hw[n] = hv;
            y += hv * xdbl[32 + n];                                     // h . C
        }
        y += Dp[d] * u[i];
        float rv = res_gate[(size_t)t * DINNER + d];
        gate[i] = y * siluf_(rv);
    }
#pragma unroll
    for (int i = 0; i < 2; ++i)
        res_gate[(size_t)t * DINNER + d0 + i] = gate[i];
}

extern "C" void kernel_launch(void* const* d_in, const int* in_sizes, int n_in,
                              void* d_out, int out_size, void* d_ws, size_t ws_size,
                              hipStream_t stream) {
    const float* inputs     = (const float*)d_in[0];
    const float* hidden     = (const float*)d_in[1];
    const float* convh      = (const float*)d_in[2];
    const float* in_proj_w  = (const float*)d_in[3];
    const float* conv_w     = (const float*)d_in[4];
    const float* conv_b     = (const float*)d_in[5];
    const float* x_proj_w   = (const float*)d_in[6];
    const float* dt_w       = (const float*)d_in[7];
    const float* dt_b       = (const float*)d_in[8];
    const float* A_log      = (const float*)d_in[9];
    const float* Dp         = (const float*)d_in[10];
    const float* out_proj_w = (const float*)d_in[11];
    const float* ln_g       = (const float*)d_in[12];
    const float* ln_b       = (const float*)d_in[13];
    const int*   rate       = (const int*)d_in[14];

    // d_out = [out (8192*256) | h (8192*512*16) | hist (8192*4*512)] fp32
    float* out     = (float*)d_out;
    float* hptr    = out  + (size_t)NTOK * DMODEL;
    float* histptr = hptr + (size_t)NTOK * DINNER * DSTATE;
    float* resg    = (float*)d_ws;                  // 8192*512 fp32 = 16 MB

    dim3 blk(256);
    // in_proj: K=256, N=1024, NT=4 -> 8 waves * 64 = 512 cols/block, grid.y=2
    gemm_wmma<DMODEL, 2 * DINNER, 4, true>
        <<<dim3(NTOK / 16, 2), blk, 0, stream>>>(inputs, in_proj_w, histptr, resg);
    // fused conv/LN/x_proj/dt/scan/gate: one block per token
    samba_step<<<dim3(NTOK), blk, 0, stream>>>(hidden, convh, conv_w, conv_b, x_proj_w,
                                               dt_w, dt_b, A_log, Dp, ln_g, ln_b, rate,
                                               hptr, histptr, resg);
    // out_proj: K=512, N=256, NT=2 -> 8 waves * 32 = 256 cols/block, grid.y=1
    gemm_wmma<DINNER, DMODEL, 2, false>
        <<<dim3(NTOK / 16, 1), blk, 0, stream>>>(resg, out_proj_w, out, nullptr);
}